// VASP_57088705299150
// MI455X (gfx1250) — compile-verified
//
#include <hip/hip_runtime.h>

// VQ-VAE nearest-codebook quantization for MI455X (gfx1250, wave32).
//   x:   (B, D, N) f32,  emb: (D, K) f32
//   out: quantized (B, D, N) f32  ++  argmin (B, N) stored as float
// Exact-fp32 WMMA (V_WMMA_F32_16X16X4_F32) GEMM for xe = x^T*emb, argmin over
// score = e2[k] - 2*xe (x^2 row term constant per row -> dropped), then a
// bandwidth-optimal LDS-staged gather.
//
// Round 5: __launch_bounds__(128, 1) on the WMMA kernel. LDS (208 KB) already
// limits the WGP to one 4-wave block (1 wave/SIMD), so let the register
// allocator use the full wave32 VGPR file instead of the 128-reg default --
// this removes the accumulator scratch spills seen in round 4.

typedef __attribute__((ext_vector_type(2))) float v2f;
typedef __attribute__((ext_vector_type(8))) float v8f;
typedef __attribute__((ext_vector_type(4))) int   v4i;

typedef __attribute__((address_space(1))) v4i* gptr_v4i;
typedef __attribute__((address_space(3))) v4i* lptr_v4i;

constexpr int Bn = 32;    // batch
constexpr int Dn = 384;   // embedding dim (contraction)
constexpr int Nn = 4096;  // latents per batch
constexpr int Kn = 512;   // codebook size

#define WAVES_PER_BLOCK 4
#define DBLK 32                      // d rows per staged emb tile
#define KGRP 128                     // codes per pass (8 k-tiles)

// ---- CDNA5 async global->LDS copy (16B), with compile-time fallback -------
__device__ __forceinline__ void cp16_g2l(float* dst_lds, const float* src) {
#if __has_builtin(__builtin_amdgcn_global_load_async_to_lds_b128)
    __builtin_amdgcn_global_load_async_to_lds_b128(
        (gptr_v4i)(const_cast<float*>(src)),
        (lptr_v4i)(dst_lds),
        /*offset=*/0, /*cpol=*/0);
#else
    *(float4*)dst_lds = *(const float4*)src;
#endif
}
__device__ __forceinline__ void cp_wait() {
#if __has_builtin(__builtin_amdgcn_global_load_async_to_lds_b128)
#if __has_builtin(__builtin_amdgcn_s_wait_asynccnt)
    __builtin_amdgcn_s_wait_asynccnt(0);
#else
    asm volatile("s_wait_asynccnt 0x0" ::: "memory");
#endif
#endif
}

// ---------------------------------------------------------------------------
// Kernel 1: e2[k] = sum_d emb[d,k]^2   (coalesced: k fastest in emb)
// ---------------------------------------------------------------------------
__global__ void vq_e2_kernel(const float* __restrict__ emb,
                             float* __restrict__ e2) {
    int k = blockIdx.x * blockDim.x + threadIdx.x;
    if (k >= Kn) return;
    float s = 0.0f;
    for (int d = 0; d < Dn; ++d) {
        float v = emb[d * Kn + k];
        s = fmaf(v, v, s);
    }
    e2[k] = s;
}

// ---------------------------------------------------------------------------
// Kernel 2: per-wave 32-row n-slab (two 16-row tiles) vs all 512 codes via
// f32 WMMA, argmin.  block = 128 threads = 4 waves.
//   dynamic LDS: 4*384*32 floats (x slabs, 192 KB) + 32*128 floats (emb tile,
//   16 KB) = 208 KB of the 320 KB WGP pool -> 1 block/WGP, 1 wave/SIMD.
// ---------------------------------------------------------------------------
__global__ void __launch_bounds__(WAVES_PER_BLOCK * 32, 1)
vq_argmin_kernel(const float* __restrict__ x,
                 const float* __restrict__ emb,
                 const float* __restrict__ e2,
                 int* __restrict__ idx_out,
                 float* __restrict__ idx_out_f) {
    extern __shared__ float xs[];                 // [4][Dn*32] ++ [DBLK*KGRP]
    float* es = xs + WAVES_PER_BLOCK * (Dn * 32); // shared emb tile

    const int lane = threadIdx.x & 31;
    const int wave = threadIdx.x >> 5;
    const int w    = blockIdx.x * WAVES_PER_BLOCK + wave;
    const int b  = w >> 7;                        // Nn/32 = 128 slabs per batch
    const int n0 = (w & 127) << 5;                // 32-row slab base

    float* xw = xs + wave * (Dn * 32);

    // ---- Stage x slab (rows d, cols n0..n0+31) into LDS: xw[d*32 + j] ----
    const float* xb = x + (size_t)b * Dn * Nn + n0;
    for (int t = lane; t < Dn * 8; t += 32) {     // 96 iters, 16B each
        int d = t >> 3, q = (t & 7) << 2;
        cp16_g2l(xw + d * 32 + q, xb + (size_t)d * Nn + q);
    }
    cp_wait();

    // WMMA lane mapping (16x16x4 f32):
    //   A: lane m = lane&15, half = lane>>4 selects K-pair {2h, 2h+1}
    //   C/D: lane column = lane&15, VGPR v -> row m = v + 8*half
    const int hf  = lane >> 4;
    const int m16 = lane & 15;

    float bestS[2][8];
    int   bestK[2][8];
#pragma unroll
    for (int t = 0; t < 2; ++t)
#pragma unroll
        for (int v = 0; v < 8; ++v) { bestS[t][v] = __builtin_inff(); bestK[t][v] = 0; }

    // 4 passes over k-groups of 128 codes; within a pass, stage 32x128 emb
    // tiles in LDS (shared by all 4 waves) and sweep D in chunks of 4.
    for (int g = 0; g < 4; ++g) {
        v8f acc0[8] = {};
        v8f acc1[8] = {};
        for (int dblk = 0; dblk < Dn / DBLK; ++dblk) {
            __syncthreads();                      // protect es from prior use
            // cooperative stage: es[r*128 + c] = emb[dblk*32 + r][g*128 + c]
            for (int t = threadIdx.x; t < (DBLK * KGRP) / 4; t += 128) {
                int r = t >> 5;                   // 32 x 16B per row
                int q = (t & 31) << 2;
                cp16_g2l(es + r * KGRP + q,
                         emb + (size_t)(dblk * DBLK + r) * Kn + g * KGRP + q);
            }
            cp_wait();
            __syncthreads();

            for (int c = 0; c < DBLK / 4; ++c) {  // 8 d-chunks of 4
                const int dl = c * 4 + 2 * hf;    // local d row (A/B K-pair)
                const float* ap = xw + (dblk * DBLK + dl) * 32 + m16;
                v2f a0, a1;
                a0.x = ap[0];  a0.y = ap[32];     // tile0 columns
                a1.x = ap[16]; a1.y = ap[48];     // tile1 columns
                const float* bp = es + dl * KGRP + m16;
                v2f bb[8];
#pragma unroll
                for (int j = 0; j < 8; ++j) {     // batch all B loads first
                    bb[j].x = bp[j * 16];
                    bb[j].y = bp[j * 16 + KGRP];
                }
#pragma unroll
                for (int j = 0; j < 8; ++j) {     // 16 WMMAs, B reused twice
                    acc0[j] = __builtin_amdgcn_wmma_f32_16x16x4_f32(
                        false, a0, false, bb[j], (short)0, acc0[j], false, false);
                    acc1[j] = __builtin_amdgcn_wmma_f32_16x16x4_f32(
                        false, a1, false, bb[j], (short)0, acc1[j], false, false);
                }
            }
        }
        // score = e2[k] - 2*xe ; update per-lane running best (8 rows/lane)
#pragma unroll
        for (int j = 0; j < 8; ++j) {
            const int k  = g * KGRP + j * 16 + m16;
            const float ek = e2[k];
#pragma unroll
            for (int v = 0; v < 8; ++v) {
                float s0 = fmaf(-2.0f, acc0[j][v], ek);
                if (s0 < bestS[0][v]) { bestS[0][v] = s0; bestK[0][v] = k; }
                float s1 = fmaf(-2.0f, acc1[j][v], ek);
                if (s1 < bestS[1][v]) { bestS[1][v] = s1; bestK[1][v] = k; }
            }
        }
    }

    // ---- Cross-lane argmin within each 16-lane half (wave32 shuffles) ----
    const size_t obase = (size_t)b * Nn + n0;
#pragma unroll
    for (int t = 0; t < 2; ++t) {
#pragma unroll
        for (int v = 0; v < 8; ++v) {
            float s = bestS[t][v];
            int   k = bestK[t][v];
#pragma unroll
            for (int off = 1; off < 16; off <<= 1) { // stays in 16-lane half
                float s2 = __shfl_xor(s, off, 32);
                int   k2 = __shfl_xor(k, off, 32);
                if (s2 < s || (s2 == s && k2 < k)) { s = s2; k = k2; }
            }
            if (m16 == v) {                        // one writer per half
                const int m = t * 16 + v + 8 * hf; // row within 32-row slab
                idx_out[obase + m]   = k;
                idx_out_f[obase + m] = (float)k;
            }
        }
    }
}

// ---------------------------------------------------------------------------
// Kernel 3: gather quantized[b,d,n] = emb[d, idx[b,n]]; fully coalesced
// stores, emb rows staged through LDS (16 KB per 8-row step).
// ---------------------------------------------------------------------------
__global__ void vq_gather_kernel(const float* __restrict__ emb,
                                 const int* __restrict__ idx,
                                 float* __restrict__ out) {
    __shared__ float er[8 * Kn];                  // 8 emb rows = 16 KB
    const int b  = blockIdx.x >> 4;               // Nn/256 = 16 chunks/batch
    const int n  = ((blockIdx.x & 15) << 8) + threadIdx.x;
    const int k  = idx[(size_t)b * Nn + n];
    float* outb  = out + (size_t)b * Dn * Nn + n;

    for (int dc = 0; dc < Dn; dc += 8) {
        __syncthreads();
        for (int t = threadIdx.x; t < 8 * Kn; t += 256)
            er[t] = emb[(size_t)dc * Kn + t];     // rows dc..dc+7, contiguous
        __syncthreads();
#pragma unroll
        for (int r = 0; r < 8; ++r)
            outb[(size_t)(dc + r) * Nn] = er[r * Kn + k];
    }
}

// ---------------------------------------------------------------------------
extern "C" void kernel_launch(void* const* d_in, const int* in_sizes, int n_in,
                              void* d_out, int out_size, void* d_ws, size_t ws_size,
                              hipStream_t stream) {
    const float* x   = (const float*)d_in[0];     // (B, D, N)
    const float* emb = (const float*)d_in[1];     // (D, K)
    float* out   = (float*)d_out;                 // quantized (B,D,N)
    float* argF  = out + (size_t)Bn * Dn * Nn;    // argmin as float (B,N)

    float* e2  = (float*)d_ws;                    // 512 floats
    int*   idx = (int*)((char*)d_ws + 4096);      // B*N ints

    // 1) codebook squared norms
    vq_e2_kernel<<<2, 256, 0, stream>>>(emb, e2);

    // 2) fp32-WMMA distance GEMM + argmin (one 32-row slab per wave)
    const int n_waves = Bn * (Nn / 32);           // 4096 waves
    const int grid2   = n_waves / WAVES_PER_BLOCK;
    const size_t lds2 =
        (size_t)(WAVES_PER_BLOCK * Dn * 32 + DBLK * KGRP) * sizeof(float); // 208 KB
    vq_argmin_kernel<<<grid2, WAVES_PER_BLOCK * 32, lds2, stream>>>(
        x, emb, e2, idx, argF);

    // 3) coalesced gather of winning codebook columns
    vq_gather_kernel<<<Bn * (Nn / 256), 256, 0, stream>>>(emb, idx, out);
}